// GAE_53575422050487
// MI455X (gfx1250) — compile-verified
//
#include <hip/hip_runtime.h>
#include <hip/hip_bf16.h>
#include <math.h>

typedef __attribute__((ext_vector_type(2))) float v2f;
typedef __attribute__((ext_vector_type(8))) float v8f;

#define GAMMA  0.99f
#define LAMDA  0.95f
#define EPS_   1e-8f

#define NB   1024      // scan blocks
#define TPB  256       // threads per block (8 wave32s)
#define RED_BLOCKS 512 // reduction blocks

// ---------------------------------------------------------------------------
// Phase 0: zero the scalar accumulator slots in workspace (graph-safe init).
// ---------------------------------------------------------------------------
__global__ void gae_init(float* acc) {
    if (threadIdx.x < 4) acc[threadIdx.x] = 0.0f;
}

// ---------------------------------------------------------------------------
// Phase 1: per-thread reverse affine aggregate (A,B):  a = B + A * carry_in.
// v[k] is cached across iterations (descending k needs v[k] and v[k+1]).
// Block composes its 256 thread aggregates into one block aggregate.
// ---------------------------------------------------------------------------
__global__ __launch_bounds__(TPB)
void gae_p1(const float* __restrict__ r, const float* __restrict__ v,
            const int* __restrict__ dn,
            float* __restrict__ thrA, float* __restrict__ thrB,
            float* __restrict__ blkA, float* __restrict__ blkB,
            long long T, long long chunk) {
    __shared__ float sA[TPB];
    __shared__ float sB[TPB];
    const int b = blockIdx.x, t = threadIdx.x;
    long long start = (long long)(b * TPB + t) * chunk;
    long long end   = start + chunk;
    if (start > T) start = T;
    if (end   > T) end   = T;

    float A = 1.0f, B = 0.0f;
    float vnext = (start < end) ? v[end] : 0.0f;   // v[k+1] for first iteration
    for (long long k = end - 1; k >= start; --k) {
        const float vk    = v[k];
        const float nd    = 1.0f - (float)dn[k];
        const float delta = fmaf(GAMMA * vnext, nd, r[k]) - vk;
        const float c     = (GAMMA * LAMDA) * nd;
        B = fmaf(c, B, delta);   // B' = delta + c*B
        A = c * A;               // A' = c*A
        vnext = vk;
    }
    thrA[b * TPB + t] = A;
    thrB[b * TPB + t] = B;
    sA[t] = A; sB[t] = B;
    __syncthreads();
    if (t == 0) {
        float Ag = 1.0f, Bg = 0.0f;
        for (int i = TPB - 1; i >= 0; --i) {  // compose right -> left
            Bg = fmaf(sA[i], Bg, sB[i]);
            Ag = sA[i] * Ag;
        }
        blkA[b] = Ag;
        blkB[b] = Bg;
    }
}

// ---------------------------------------------------------------------------
// Phase 2: serial reverse composition over NB block aggregates -> block carries.
// ---------------------------------------------------------------------------
__global__ void gae_p2(const float* __restrict__ blkA, const float* __restrict__ blkB,
                       float* __restrict__ carry) {
    if (blockIdx.x == 0 && threadIdx.x == 0) {
        float x = 0.0f;
        for (int b = NB - 1; b >= 0; --b) {
            carry[b] = x;
            x = fmaf(blkA[b], x, blkB[b]);
        }
    }
}

// ---------------------------------------------------------------------------
// Phase 3: rescan with true carries; write unnormalized advantages.
// ---------------------------------------------------------------------------
__global__ __launch_bounds__(TPB)
void gae_p3(const float* __restrict__ r, const float* __restrict__ v,
            const int* __restrict__ dn,
            const float* __restrict__ thrA, const float* __restrict__ thrB,
            const float* __restrict__ carry,
            float* __restrict__ out, long long T, long long chunk) {
    __shared__ float tc[TPB];
    const int b = blockIdx.x, t = threadIdx.x;
    if (t == 0) {
        float x = carry[b];
        for (int i = TPB - 1; i >= 0; --i) {
            tc[i] = x;
            x = fmaf(thrA[b * TPB + i], x, thrB[b * TPB + i]);
        }
    }
    __syncthreads();

    long long start = (long long)(b * TPB + t) * chunk;
    long long end   = start + chunk;
    if (start > T) start = T;
    if (end   > T) end   = T;

    float x = tc[t];
    float vnext = (start < end) ? v[end] : 0.0f;
    for (long long k = end - 1; k >= start; --k) {
        const float vk    = v[k];
        const float nd    = 1.0f - (float)dn[k];
        const float delta = fmaf(GAMMA * vnext, nd, r[k]) - vk;
        const float c     = (GAMMA * LAMDA) * nd;
        x = fmaf(c, x, delta);
        out[k] = x;
        vnext = vk;
    }
}

// ---------------------------------------------------------------------------
// Phase 4: WMMA-based sum / sum-of-squares reduction over full 64-elem tiles.
// Each wave step: one global_load_b64 per lane (float2 -> A 16x4 f32 operand),
// B = all-ones 4x16, accumulate into v8f C tiles. No masking in the hot loop:
// only Tmain = T & ~63 is covered here; the <64-elem tail is folded serially
// into gae_stats. C-tile grand total == 16 * sum(elements).
// ---------------------------------------------------------------------------
__global__ __launch_bounds__(TPB)
void gae_reduce(const float* __restrict__ x,
                float* __restrict__ partS, float* __restrict__ partQ,
                long long Tmain) {
    __shared__ float ls[TPB / 32];
    __shared__ float lq[TPB / 32];
    const int lane   = threadIdx.x & 31;
    const int waveIn = threadIdx.x >> 5;
    const int wave   = (int)(((long long)blockIdx.x * blockDim.x + threadIdx.x) >> 5);
    const int nwaves = (int)(((long long)gridDim.x * blockDim.x) >> 5);

    const float2* __restrict__ xv = (const float2*)x;
    const int npair = (int)(Tmain >> 1);   // pairs of f32 (fits in int for T<2^31)

    v8f cs = {};
    v8f cq = {};
    v2f ones; ones.x = 1.0f; ones.y = 1.0f;

    for (int p = wave * 32; p < npair; p += nwaves * 32) {
        const float2 a = xv[p + lane];     // single b64 load per lane
        v2f av;  av.x = a.x;          av.y = a.y;
        v2f av2; av2.x = a.x * a.x;   av2.y = a.y * a.y;
        cs = __builtin_amdgcn_wmma_f32_16x16x4_f32(
                false, av,  false, ones, (short)0, cs, false, false);
        cq = __builtin_amdgcn_wmma_f32_16x16x4_f32(
                false, av2, false, ones, (short)0, cq, false, false);
    }

    float s = 0.0f, q = 0.0f;
#pragma unroll
    for (int i = 0; i < 8; ++i) { s += cs[i]; q += cq[i]; }
#pragma unroll
    for (int off = 16; off >= 1; off >>= 1) {
        s += __shfl_xor(s, off, 32);
        q += __shfl_xor(q, off, 32);
    }
    if (lane == 0) { ls[waveIn] = s * (1.0f / 16.0f); lq[waveIn] = q * (1.0f / 16.0f); }
    __syncthreads();
    if (threadIdx.x == 0) {
        float bs = 0.0f, bq = 0.0f;
        for (int w = 0; w < TPB / 32; ++w) { bs += ls[w]; bq += lq[w]; }
        partS[blockIdx.x] = bs;
        partQ[blockIdx.x] = bq;
    }
}

// ---------------------------------------------------------------------------
// Phase 5: final stats. Deterministic serial sum of per-block partials plus
// the (<64-element) tail not covered by the WMMA tiles.
// ---------------------------------------------------------------------------
__global__ void gae_stats(const float* __restrict__ partS,
                          const float* __restrict__ partQ,
                          const float* __restrict__ x,
                          float* __restrict__ acc,
                          long long Tmain, long long T) {
    if (blockIdx.x == 0 && threadIdx.x == 0) {
        float s = 0.0f, q = 0.0f;
        for (int i = 0; i < RED_BLOCKS; ++i) { s += partS[i]; q += partQ[i]; }
        for (long long k = Tmain; k < T; ++k) {   // tail (empty when T % 64 == 0)
            const float y = x[k];
            s += y; q += y * y;
        }
        const float mean = s / (float)T;
        float var = q / (float)T - mean * mean;
        if (var < 0.0f) var = 0.0f;
        acc[2] = mean;
        acc[3] = 1.0f / (sqrtf(var) + EPS_);
    }
}

// ---------------------------------------------------------------------------
// Phase 6: standardize in place, float4-vectorized.
// ---------------------------------------------------------------------------
__global__ __launch_bounds__(TPB)
void gae_norm(float* __restrict__ out, const float* __restrict__ acc, long long T) {
    const float mean = acc[2];
    const float inv  = acc[3];
    const long long tid    = (long long)blockIdx.x * blockDim.x + threadIdx.x;
    const long long stride = (long long)gridDim.x * blockDim.x;

    float4* __restrict__ o4 = (float4*)out;
    const long long nvec = T >> 2;
    for (long long i = tid; i < nvec; i += stride) {
        float4 y = o4[i];
        y.x = (y.x - mean) * inv;
        y.y = (y.y - mean) * inv;
        y.z = (y.z - mean) * inv;
        y.w = (y.w - mean) * inv;
        o4[i] = y;
    }
    for (long long k = (nvec << 2) + tid; k < T; k += stride)   // tail
        out[k] = (out[k] - mean) * inv;
}

// ---------------------------------------------------------------------------
extern "C" void kernel_launch(void* const* d_in, const int* in_sizes, int n_in,
                              void* d_out, int out_size, void* d_ws, size_t ws_size,
                              hipStream_t stream) {
    const float* r  = (const float*)d_in[0];   // rewards [T]
    const float* v  = (const float*)d_in[1];   // v_pred  [T+1]
    const int*   dn = (const int*)d_in[2];     // dones   [T]
    float* out = (float*)d_out;
    const long long T = (long long)in_sizes[0];
    const long long Tmain = T & ~63LL;         // full 64-element WMMA tiles

    // Workspace layout (floats)
    float* ws    = (float*)d_ws;
    float* acc   = ws;                         // [0..3]: sum,sumsq,mean,inv
    float* blkA  = ws + 4;                     // NB
    float* blkB  = blkA + NB;                  // NB
    float* carry = blkB + NB;                  // NB
    float* partS = carry + NB;                 // RED_BLOCKS
    float* partQ = partS + RED_BLOCKS;         // RED_BLOCKS
    float* thrA  = partQ + RED_BLOCKS;         // NB*TPB
    float* thrB  = thrA + (long long)NB * TPB; // NB*TPB

    const long long nthreads = (long long)NB * TPB;
    const long long chunk = (T + nthreads - 1) / nthreads;  // contiguous elems/thread

    gae_init  <<<1, 64, 0, stream>>>(acc);
    gae_p1    <<<NB, TPB, 0, stream>>>(r, v, dn, thrA, thrB, blkA, blkB, T, chunk);
    gae_p2    <<<1, 1, 0, stream>>>(blkA, blkB, carry);
    gae_p3    <<<NB, TPB, 0, stream>>>(r, v, dn, thrA, thrB, carry, out, T, chunk);
    gae_reduce<<<RED_BLOCKS, TPB, 0, stream>>>(out, partS, partQ, Tmain);
    gae_stats <<<1, 1, 0, stream>>>(partS, partQ, out, acc, Tmain, T);
    gae_norm  <<<2048, TPB, 0, stream>>>(out, acc, T);
}